// MultiHeadedAttention_62354335203789
// MI455X (gfx1250) — compile-verified
//
#include <hip/hip_runtime.h>
#include <hip/hip_bf16.h>

typedef float v2f __attribute__((ext_vector_type(2)));
typedef float v8f __attribute__((ext_vector_type(8)));
typedef int   v4i __attribute__((vector_size(16)));   // GCC-style, matches builtin

#define SEQ    2048
#define HID    1024
#define NHEAD  16
#define DHEAD  64
#define BATCH  2
#define ROWS   (BATCH * SEQ)   // 4096

#define AS1 __attribute__((address_space(1)))
#define AS3 __attribute__((address_space(3)))

__device__ __forceinline__ v8f wmma_f32(v2f a, v2f b, v8f c) {
    // V_WMMA_F32_16X16X4_F32 : D = A(16x4,f32) * B(4x16,f32) + C(16x16,f32)
    return __builtin_amdgcn_wmma_f32_16x16x4_f32(
        false, a, false, b, (short)0, c, false, false);
}

// 16-byte global -> LDS async copy (ASYNCcnt path)
__device__ __forceinline__ void async_copy16(void* lds_dst, const void* gsrc) {
#if __has_builtin(__builtin_amdgcn_global_load_async_to_lds_b128)
    __builtin_amdgcn_global_load_async_to_lds_b128(
        (AS1 v4i*)gsrc, (AS3 v4i*)lds_dst, 0, 0);
#else
    unsigned lds32 = (unsigned)(unsigned long long)(AS3 char*)lds_dst;
    asm volatile("global_load_async_to_lds_b128 %0, %1, off"
                 :: "v"(lds32), "v"(gsrc) : "memory");
#endif
}

template <int N>
__device__ __forceinline__ void wait_async() {
#if __has_builtin(__builtin_amdgcn_s_wait_asynccnt)
    __builtin_amdgcn_s_wait_asynccnt(N);
#else
    asm volatile("s_wait_asynccnt %0" :: "i"(N) : "memory");
#endif
}

// ---------------------------------------------------------------------------
// Shared GEMM core: block = 256 thr (8 waves). Block tile: M=32, N=512.
// A panel (32 x 64-K chunk) staged in LDS via async copies, double-buffered,
// shared by all 8 waves.  Each wave: 2 m-subtiles x 4 n-subtiles (16x64 each).
// ---------------------------------------------------------------------------
__device__ __forceinline__ void stage_A(const float* __restrict__ A, int m0,
                                        int k0, float (&buf)[32][68], int t) {
#pragma unroll
    for (int s = 0; s < 2; ++s) {
        const int flat = t + 256 * s;       // 0..511
        const int row  = flat >> 4;         // 0..31
        const int seg  = flat & 15;         // 0..15 (4 floats each)
        async_copy16(&buf[row][seg * 4],
                     A + (size_t)(m0 + row) * HID + k0 + seg * 4);
    }
}

__device__ __forceinline__ void gemm_block_core(const float* __restrict__ A,
                                                const float* __restrict__ W,
                                                int m0, int n0,
                                                float (&abuf)[2][32][68],
                                                v8f (&acc)[2][4]) {
    const int t    = threadIdx.x;
    const int lane = t & 31;
    const int half = lane >> 4;
    const int l16  = lane & 15;

#pragma unroll
    for (int mt = 0; mt < 2; ++mt)
#pragma unroll
        for (int nt = 0; nt < 4; ++nt)
#pragma unroll
            for (int i = 0; i < 8; ++i) acc[mt][nt][i] = 0.0f;

    const float* wr[4];
#pragma unroll
    for (int nt = 0; nt < 4; ++nt)
        wr[nt] = W + (size_t)(n0 + nt * 16 + l16) * HID + half * 2;

    stage_A(A, m0, 0, abuf[0], t);          // 2 async ops in flight

    for (int kc = 0; kc < HID / 64; ++kc) {
        if (kc + 1 < HID / 64) {
            stage_A(A, m0, (kc + 1) * 64, abuf[(kc + 1) & 1], t);
            wait_async<2>();                // chunk kc landed, kc+1 in flight
        } else {
            wait_async<0>();
        }
        __syncthreads();

        const float(&buf)[32][68] = abuf[kc & 1];
        const int kbase = kc * 64;
#pragma unroll 4
        for (int kk = 0; kk < 16; ++kk) {
            v2f a0 = *(const v2f*)&buf[l16][kk * 4 + half * 2];
            v2f a1 = *(const v2f*)&buf[16 + l16][kk * 4 + half * 2];
#pragma unroll
            for (int nt = 0; nt < 4; ++nt) {
                v2f bf;
                bf.x = wr[nt][kbase + kk * 4];
                bf.y = wr[nt][kbase + kk * 4 + 1];
                acc[0][nt] = wmma_f32(a0, bf, acc[0][nt]);
                acc[1][nt] = wmma_f32(a1, bf, acc[1][nt]);
            }
        }
        __syncthreads();                    // all waves done with this buffer
    }
}

// ---------------------------------------------------------------------------
// Kernel 1: QKV projection.  grid = (ROWS/32, 3072/512) = (128, 6), block 256
// ---------------------------------------------------------------------------
__global__ void qkv_kernel(const float* __restrict__ X,
                           const float* __restrict__ Wq,
                           const float* __restrict__ bq,
                           float* __restrict__ Qb,
                           float* __restrict__ Kb,
                           float* __restrict__ Vb) {
    __shared__ float abuf[2][32][68];
    const int t    = threadIdx.x;
    const int lane = t & 31;
    const int w    = t >> 5;
    const int half = lane >> 4;
    const int l16  = lane & 15;
    const int m0   = blockIdx.x * 32;
    const int n0   = (blockIdx.y * 8 + w) * 64;

    v8f acc[2][4];
    gemm_block_core(X, Wq, m0, n0, abuf, acc);

#pragma unroll
    for (int mt = 0; mt < 2; ++mt)
#pragma unroll
        for (int nt = 0; nt < 4; ++nt) {
            const int col   = n0 + nt * 16 + l16;   // o in [0,3072)
            const int which = col >> 10;
            const int head  = (col >> 6) & 15;
            const int d     = col & 63;
            const float bias = bq[col];
            float* dst = (which == 0) ? Qb : ((which == 1) ? Kb : Vb);
#pragma unroll
            for (int i = 0; i < 8; ++i) {
                const int row = m0 + mt * 16 + half * 8 + i;  // b*SEQ+s
                const int bb  = row >> 11;
                const int s   = row & (SEQ - 1);
                dst[(((size_t)(bb * NHEAD + head)) * SEQ + s) * DHEAD + d] =
                    acc[mt][nt][i] + bias;
            }
        }
}

// ---------------------------------------------------------------------------
// Kernel 2: flash attention.  grid = (16, 32), block 256 (8 waves).
// All 8 waves share (b,h): K/V tiles staged in LDS (async, double-buffered).
// ---------------------------------------------------------------------------
__global__ void attn_kernel(const float* __restrict__ Q,
                            const float* __restrict__ K,
                            const float* __restrict__ V,
                            float* __restrict__ ctx) {
    __shared__ float kbuf[2][16][68];
    __shared__ float vbuf[2][16][68];
    __shared__ float plds[8][16][17];

    const int t    = threadIdx.x;
    const int lane = t & 31;
    const int w    = t >> 5;
    const int half = lane >> 4;
    const int l16  = lane & 15;
    const int bh   = blockIdx.y;
    const int b    = bh >> 4;
    const int h    = bh & 15;
    const int q0   = (blockIdx.x * 8 + w) * 16;
    const int srow = t >> 4;                 // staging row 0..15
    const int sseg = t & 15;                 // staging 16B segment

    const size_t base = (size_t)bh * SEQ * DHEAD;
    const float* Qp = Q + base;
    const float* Kp = K + base;
    const float* Vp = V + base;

    // Q A-fragments for the whole head dim (16 K-steps of 4)
    v2f qa[16];
    {
        const float* qrow = Qp + (size_t)(q0 + l16) * DHEAD + half * 2;
#pragma unroll
        for (int kk = 0; kk < 16; ++kk) {
            qa[kk].x = qrow[kk * 4 + 0];
            qa[kk].y = qrow[kk * 4 + 1];
        }
    }

    float m_r[8], l_r[8];
    v8f accO[4];
#pragma unroll
    for (int i = 0; i < 8; ++i) { m_r[i] = -3.0e38f; l_r[i] = 0.0f; }
#pragma unroll
    for (int nt = 0; nt < 4; ++nt)
#pragma unroll
        for (int i = 0; i < 8; ++i) accO[nt][i] = 0.0f;

    // stage K/V tile 0
    async_copy16(&kbuf[0][srow][sseg * 4], Kp + (size_t)srow * DHEAD + sseg * 4);
    async_copy16(&vbuf[0][srow][sseg * 4], Vp + (size_t)srow * DHEAD + sseg * 4);

    for (int j = 0; j < SEQ / 16; ++j) {
        const int pb = j & 1;
        if (j + 1 < SEQ / 16) {
            const size_t off = (size_t)((j + 1) * 16 + srow) * DHEAD + sseg * 4;
            async_copy16(&kbuf[pb ^ 1][srow][sseg * 4], Kp + off);
            async_copy16(&vbuf[pb ^ 1][srow][sseg * 4], Vp + off);
            wait_async<2>();                 // tile j landed, j+1 in flight
        } else {
            wait_async<0>();
        }
        __syncthreads();                     // staging visible to all waves

        // ---- S tile (16x16) = Q * K_j^T (K from LDS) -----------------------
        v8f sacc;
#pragma unroll
        for (int i = 0; i < 8; ++i) sacc[i] = 0.0f;
#pragma unroll
        for (int kk = 0; kk < 16; ++kk) {
            v2f bf = *(const v2f*)&kbuf[pb][l16][kk * 4 + half * 2];
            sacc = wmma_f32(qa[kk], bf, sacc);
        }

        // ---- online softmax ------------------------------------------------
        float p8[8], esc[8];
#pragma unroll
        for (int i = 0; i < 8; ++i) {
            float s = sacc[i] * 0.125f;      // 1/sqrt(64)
            float v = s;
            v = fmaxf(v, __shfl_xor(v, 1));
            v = fmaxf(v, __shfl_xor(v, 2));
            v = fmaxf(v, __shfl_xor(v, 4));
            v = fmaxf(v, __shfl_xor(v, 8));
            const float mn = fmaxf(m_r[i], v);
            const float p  = __expf(s - mn);
            float rs = p;
            rs += __shfl_xor(rs, 1);
            rs += __shfl_xor(rs, 2);
            rs += __shfl_xor(rs, 4);
            rs += __shfl_xor(rs, 8);
            esc[i] = __expf(m_r[i] - mn);
            l_r[i] = l_r[i] * esc[i] + rs;
            m_r[i] = mn;
            p8[i]  = p;
        }

        // ---- P : C-layout -> A-layout via LDS ------------------------------
#pragma unroll
        for (int i = 0; i < 8; ++i) plds[w][half * 8 + i][l16] = p8[i];
        __syncthreads();                     // P visible

#pragma unroll
        for (int nt = 0; nt < 4; ++nt)
#pragma unroll
            for (int i = 0; i < 8; ++i) accO[nt][i] *= esc[i];

        // ---- accO (16x64) += P (16x16) * V_j (V from LDS) ------------------
#pragma unroll
        for (int kk = 0; kk < 4; ++kk) {
            v2f af;
            af.x = plds[w][l16][kk * 4 + half * 2 + 0];
            af.y = plds[w][l16][kk * 4 + half * 2 + 1];
            const int kr = kk * 4 + half * 2;
#pragma unroll
            for (int nt = 0; nt < 4; ++nt) {
                v2f bf;
                bf.x = vbuf[pb][kr + 0][nt * 16 + l16];
                bf.y = vbuf[pb][kr + 1][nt * 16 + l16];
                accO[nt] = wmma_f32(af, bf, accO[nt]);
            }
        }
        __syncthreads();   // reads of pb buffers + plds done before overwrite
    }

#pragma unroll
    for (int i = 0; i < 8; ++i) {
        const float inv = 1.0f / l_r[i];
        const int sr = q0 + half * 8 + i;
        float* orow = ctx + ((size_t)(b * SEQ + sr)) * HID + h * DHEAD;
#pragma unroll
        for (int nt = 0; nt < 4; ++nt)
            orow[nt * 16 + l16] = accO[nt][i] * inv;
    }
}

// ---------------------------------------------------------------------------
// Kernel 3: output projection.  grid = (ROWS/32, 1024/512) = (128, 2)
// ---------------------------------------------------------------------------
__global__ void proj_kernel(const float* __restrict__ Cx,
                            const float* __restrict__ Wo,
                            const float* __restrict__ bo,
                            float* __restrict__ Out) {
    __shared__ float abuf[2][32][68];
    const int t    = threadIdx.x;
    const int lane = t & 31;
    const int w    = t >> 5;
    const int half = lane >> 4;
    const int l16  = lane & 15;
    const int m0   = blockIdx.x * 32;
    const int n0   = (blockIdx.y * 8 + w) * 64;

    v8f acc[2][4];
    gemm_block_core(Cx, Wo, m0, n0, abuf, acc);

#pragma unroll
    for (int mt = 0; mt < 2; ++mt)
#pragma unroll
        for (int nt = 0; nt < 4; ++nt) {
            const int col = n0 + nt * 16 + l16;
            const float bias = bo[col];
#pragma unroll
            for (int i = 0; i < 8; ++i) {
                const int row = m0 + mt * 16 + half * 8 + i;
                Out[(size_t)row * HID + col] = acc[mt][nt][i] + bias;
            }
        }
}

// ---------------------------------------------------------------------------
extern "C" void kernel_launch(void* const* d_in, const int* in_sizes, int n_in,
                              void* d_out, int out_size, void* d_ws, size_t ws_size,
                              hipStream_t stream) {
    const float* x     = (const float*)d_in[0];
    const float* w_qkv = (const float*)d_in[1];
    const float* b_qkv = (const float*)d_in[2];
    const float* w_out = (const float*)d_in[3];
    const float* b_out = (const float*)d_in[4];
    float* out = (float*)d_out;

    // workspace: Q, K, V as (B, HEADS, S, D), then ctx as (B, S, HID)
    float* Qb  = (float*)d_ws;
    float* Kb  = Qb + (size_t)ROWS * HID;
    float* Vb  = Kb + (size_t)ROWS * HID;
    float* ctx = Vb + (size_t)ROWS * HID;

    qkv_kernel <<<dim3(ROWS / 32, 6),      256, 0, stream>>>(x, w_qkv, b_qkv, Qb, Kb, Vb);
    attn_kernel<<<dim3(16, BATCH * NHEAD), 256, 0, stream>>>(Qb, Kb, Vb, ctx);
    proj_kernel<<<dim3(ROWS / 32, 2),      256, 0, stream>>>(ctx, w_out, b_out, out);
}